// HardConstrainedMLP_unroll_50053548867652
// MI455X (gfx1250) — compile-verified
//
#include <hip/hip_runtime.h>

// Problem constants (reference): B=4096, D_IN=128, HID=200, DIM=256, M=32
// SIGMA*OMEGA = 1.7, box = [0,1]

typedef __attribute__((ext_vector_type(2))) float v2f;
typedef __attribute__((ext_vector_type(8))) float v8f;

__device__ __forceinline__ v8f wmma_f32(v2f a, v2f b, v8f c) {
  // D = A(16x4) * B(4x16) + C(16x16), fp32 matrix pipe
  return __builtin_amdgcn_wmma_f32_16x16x4_f32(
      /*neg_a=*/false, a, /*neg_b=*/false, b,
      /*c_mod=*/(short)0, c, /*reuse_a=*/false, /*reuse_b=*/false);
}

// ---------------------------------------------------------------------------
// Kernel 1: H = inv(A A^T + 1e-6 I) via Gauss-Jordan (32x32, SPD), then
//           Aprime = H @ A  [32 x 256]  -> workspace
// ---------------------------------------------------------------------------
__global__ void prep_kernel(const float* __restrict__ A, float* __restrict__ Aprime) {
  __shared__ float aug[32][65];  // [S | I]
  const int tid = threadIdx.x;   // 256 threads

  for (int e = tid; e < 32 * 32; e += 256) {
    int i = e >> 5, j = e & 31;
    float s = 0.f;
    for (int k = 0; k < 256; ++k) s += A[i * 256 + k] * A[j * 256 + k];
    if (i == j) s += 1e-6f;
    aug[i][j] = s;
    aug[i][32 + j] = (i == j) ? 1.f : 0.f;
  }
  __syncthreads();

  for (int p = 0; p < 32; ++p) {
    __syncthreads();
    float piv = aug[p][p];
    __syncthreads();
    if (tid < 64) aug[p][tid] *= (1.0f / piv);
    __syncthreads();
    int i = tid >> 3;              // 32 rows, 8 threads per row
    float f = aug[i][p];
    __syncthreads();
    if (i != p) {
      int j0 = (tid & 7) << 3;
      for (int jj = 0; jj < 8; ++jj) {
        int j = j0 + jj;
        aug[i][j] -= f * aug[p][j];
      }
    }
  }
  __syncthreads();

  // Aprime[i][n] = sum_j H[i][j] * A[j][n]
  for (int e = tid; e < 32 * 256; e += 256) {
    int i = e >> 8, n = e & 255;
    float s = 0.f;
    for (int j = 0; j < 32; ++j) s += aug[i][32 + j] * A[j * 256 + n];
    Aprime[i * 256 + n] = s;
  }
}

// ---------------------------------------------------------------------------
// Kernel 2: out[16x16 tile] = act(in[Bt x K] @ W[K x N] + bias), fp32 WMMA.
// One wave per output tile. K is always a multiple of 4 (128 or 200).
// ---------------------------------------------------------------------------
__global__ void mlp_gemm_kernel(const float* __restrict__ in, const float* __restrict__ W,
                                const float* __restrict__ bias, float* __restrict__ out,
                                int K, int N, int relu) {
  const int lane = threadIdx.x & 31;
  const int mn = lane & 15;   // A-op row / B-op col within tile
  const int h = lane >> 4;    // K-half selector
  const int bm = blockIdx.x * 16;
  const int bn = blockIdx.y * 16;
  const int col = bn + mn;
  const bool colOK = (col < N);
  const int colc = colOK ? col : (N - 1);

  v8f acc;
  float bv = colOK ? bias[colc] : 0.f;
#pragma unroll
  for (int v = 0; v < 8; ++v) acc[v] = bv;

  const float* arow = in + (bm + mn) * K;
  for (int k = 0; k < K; k += 4) {
    v2f a = *(const v2f*)(arow + k + 2 * h);   // 8B aligned (K even, k+2h even)
    int kr = k + 2 * h;
    v2f b;
    float w0 = W[kr * N + colc];
    float w1 = W[(kr + 1) * N + colc];
    b[0] = colOK ? w0 : 0.f;
    b[1] = colOK ? w1 : 0.f;
    acc = wmma_f32(a, b, acc);
  }

  if (relu) {
#pragma unroll
    for (int v = 0; v < 8; ++v) acc[v] = fmaxf(acc[v], 0.f);
  }
  if (colOK) {
#pragma unroll
    for (int v = 0; v < 8; ++v) out[(bm + v + 8 * h) * N + col] = acc[v];
  }
}

// ---------------------------------------------------------------------------
// Kernel 3: Douglas-Rachford iteration. One 16-row tile per WG; TWO waves
// cooperate on the tile:
//   - each wave owns 8 of the 16 N-tiles of z (64 VGPRs) and of c (64 VGPRs)
//   - GEMM1 r = z @ A^T is split K-wise (each wave K=128 -> partial r in LDS,
//     summed when loading A-fragments for GEMM2)
//   - GEMM2 d = r @ A' is split across N-tiles (8 per wave), accumulator is
//     initialized with -c so u = z - d directly.
// A^T and A' are staged in LDS in a row-pair-interleaved layout so each
// B-operand fragment is one ds_load_b64 (pair strides = 32 mod 64 banks ->
// the two lane halves use disjoint bank halves).
// ---------------------------------------------------------------------------
#define ATP 96    // A^T pair-row stride (128 pair-rows x 64 cols)
#define APP 544   // A' pair-row stride  (16 pair-rows x 512 cols)
#define ZS  260   // z-buffer row stride (16 x 256)
#define RS  36    // r partial-buffer row stride (16 x 32)

__global__ void dr_iter_kernel(const float* __restrict__ y, const float* __restrict__ bmat,
                               const float* __restrict__ A, const float* __restrict__ Aprime,
                               const int* __restrict__ n_iter_p, float* __restrict__ out) {
  extern __shared__ float smem[];
  float* At2   = smem;                  // 128*ATP = 12288
  float* Ap2   = At2 + 128 * ATP;       // 16*APP  = 8704
  float* zbuf  = Ap2 + 16 * APP;        // 16*ZS   = 4160
  float* rbufs = zbuf + 16 * ZS;        // 2*16*RS = 1152

  const int tid = threadIdx.x;          // 64 threads = 2 waves
  const int lane = tid & 31;
  const int wave = tid >> 5;
  const int mn = lane & 15;
  const int h = lane >> 4;
  const int rowbase = blockIdx.x * 16;
  const int n_iter = n_iter_p[0];
  float* rb_mine = rbufs + wave * (16 * RS);

  // Stage A^T and A' into pair-interleaved LDS: elem (k,n) at [(k>>1)*S + n*2 + (k&1)]
  for (int e = tid; e < 256 * 32; e += 64) {
    int k = e >> 5, n = e & 31;
    At2[(k >> 1) * ATP + n * 2 + (k & 1)] = A[n * 256 + k];
  }
  for (int e = tid; e < 32 * 256; e += 64) {
    int k = e >> 8, n = e & 255;
    Ap2[(k >> 1) * APP + n * 2 + (k & 1)] = Aprime[e];
  }
  // b tile (16x32) staged in rbufs[0] region (stride RS) for c = b @ A'
  for (int i = tid; i < 16 * 32; i += 64) {
    int r = i >> 5, c = i & 31;
    rbufs[r * RS + c] = bmat[(rowbase + r) * 32 + c];
  }
  __syncthreads();

  // ncarr[j] = -(b @ A') for this wave's 8 N-tiles; kept in registers.
  v2f av[8];
#pragma unroll
  for (int j = 0; j < 8; ++j)
    av[j] = *(const v2f*)(rbufs + mn * RS + 4 * j + 2 * h);

  v8f ncarr[8];
#pragma unroll
  for (int j = 0; j < 8; ++j) {
    const int nb = (wave * 8 + j) * 16;
    v8f acc = {};
#pragma unroll
    for (int kk = 0; kk < 8; ++kk) {
      v2f bb = *(const v2f*)(Ap2 + (2 * kk + h) * APP + (nb + mn) * 2);
      acc = wmma_f32(av[kk], bb, acc);
    }
#pragma unroll
    for (int v = 0; v < 8; ++v) ncarr[j][v] = -acc[v];
  }

  // z0 = y tile (this wave's 8 N-tiles), C/D layout in registers
  v8f z[8];
#pragma unroll
  for (int j = 0; j < 8; ++j) {
#pragma unroll
    for (int v = 0; v < 8; ++v)
      z[j][v] = y[(rowbase + v + 8 * h) * 256 + (wave * 8 + j) * 16 + mn];
  }

  const int k0 = wave * 128;  // this wave's K-half of GEMM1

  for (int t = 0; t <= n_iter; ++t) {
    // z regs -> shared zbuf (both waves contribute their column halves)
#pragma unroll
    for (int j = 0; j < 8; ++j) {
#pragma unroll
      for (int v = 0; v < 8; ++v)
        zbuf[(v + 8 * h) * ZS + (wave * 8 + j) * 16 + mn] = z[j][v];
    }
    __syncthreads();

    // GEMM1 partial: r_w = z[:, k0:k0+128] @ A^T[k0:k0+128, :]
    v8f r0 = {}, r1 = {};
#pragma unroll 4
    for (int k = k0; k < k0 + 128; k += 4) {
      v2f a = *(const v2f*)(zbuf + mn * ZS + k + 2 * h);
      const int pr = (k >> 1) + h;  // pair-row of rows (k+2h, k+2h+1)
      v2f b0 = *(const v2f*)(At2 + pr * ATP + mn * 2);
      v2f b1 = *(const v2f*)(At2 + pr * ATP + (16 + mn) * 2);
      r0 = wmma_f32(a, b0, r0);
      r1 = wmma_f32(a, b1, r1);
    }
#pragma unroll
    for (int v = 0; v < 8; ++v) {
      rb_mine[(v + 8 * h) * RS + mn] = r0[v];
      rb_mine[(v + 8 * h) * RS + 16 + mn] = r1[v];
    }
    __syncthreads();

    // Sum the two partials while hoisting the 8 A-operand fragments of r.
#pragma unroll
    for (int j = 0; j < 8; ++j) {
      const int off = mn * RS + 4 * j + 2 * h;
      v2f p0 = *(const v2f*)(rbufs + off);
      v2f p1 = *(const v2f*)(rbufs + 16 * RS + off);
      av[j] = p0 + p1;
    }

    const bool last = (t == n_iter);
    // GEMM2 + fused update on this wave's 8 N-tiles; acc starts at -c.
#pragma unroll
    for (int j = 0; j < 8; ++j) {
      const int nb = (wave * 8 + j) * 16;
      v8f d = ncarr[j];
#pragma unroll
      for (int kk = 0; kk < 8; ++kk) {
        v2f bb = *(const v2f*)(Ap2 + (2 * kk + h) * APP + (nb + mn) * 2);
        d = wmma_f32(av[kk], bb, d);
      }
#pragma unroll
      for (int v = 0; v < 8; ++v) {
        float u = z[j][v] - d[v];          // = z - r@A' + c
        if (last) {
          out[(rowbase + v + 8 * h) * 256 + nb + mn] = u;
        } else {
          float refl = 2.f * u - z[j][v];
          float pb = fminf(fmaxf(refl, 0.f), 1.f);
          z[j][v] = z[j][v] + 1.7f * (pb - u);
        }
      }
    }
  }
}

// ---------------------------------------------------------------------------
extern "C" void kernel_launch(void* const* d_in, const int* in_sizes, int n_in,
                              void* d_out, int out_size, void* d_ws, size_t ws_size,
                              hipStream_t stream) {
  (void)in_sizes; (void)n_in; (void)out_size; (void)ws_size;
  const float* x  = (const float*)d_in[0];
  const float* b  = (const float*)d_in[1];
  const float* W1 = (const float*)d_in[2];
  const float* b1 = (const float*)d_in[3];
  const float* W2 = (const float*)d_in[4];
  const float* b2 = (const float*)d_in[5];
  const float* W3 = (const float*)d_in[6];
  const float* b3 = (const float*)d_in[7];
  const float* A  = (const float*)d_in[8];
  const int* n_iter = (const int*)d_in[10];
  float* out = (float*)d_out;

  float* ws = (float*)d_ws;
  float* Aprime = ws;                    // 32*256
  float* h1 = ws + 32 * 256;             // 4096*200
  float* h2 = h1 + 4096 * 200;           // 4096*200
  float* y  = h2 + 4096 * 200;           // 4096*256

  prep_kernel<<<1, 256, 0, stream>>>(A, Aprime);

  // MLP: 4096 rows, one wave per 16x16 tile
  mlp_gemm_kernel<<<dim3(256, 13), 32, 0, stream>>>(x,  W1, b1, h1, 128, 200, 1);
  mlp_gemm_kernel<<<dim3(256, 13), 32, 0, stream>>>(h1, W2, b2, h2, 200, 200, 1);
  mlp_gemm_kernel<<<dim3(256, 16), 32, 0, stream>>>(h2, W3, b3, y,  200, 256, 0);

  size_t smem_bytes = (size_t)(128 * ATP + 16 * APP + 16 * ZS + 2 * 16 * RS) * sizeof(float);
  dr_iter_kernel<<<256, 64, smem_bytes, stream>>>(y, b, A, Aprime, n_iter, out);
}